// GNNModel_30365418783238
// MI455X (gfx1250) — compile-verified
//
#include <hip/hip_runtime.h>

typedef __attribute__((ext_vector_type(2))) float v2f;
typedef __attribute__((ext_vector_type(8))) float v8f;

#define BN_EPS 1e-5f

// ---------------------------------------------------------------- utilities

__global__ void fill_f32(float* __restrict__ p, float v, long n) {
  long i = (long)blockIdx.x * blockDim.x + threadIdx.x;
  long stride = (long)gridDim.x * blockDim.x;
  for (; i < n; i += stride) p[i] = v;
}

// deg[row[e]] += 1  (deg pre-initialized to 1.0 == self loop contribution)
__global__ void degree_kernel(const int* __restrict__ row, float* __restrict__ deg, int E) {
  int i = blockIdx.x * blockDim.x + threadIdx.x;
  int stride = gridDim.x * blockDim.x;
  for (; i < E; i += stride)
    __hip_atomic_fetch_add(&deg[row[i]], 1.0f, __ATOMIC_RELAXED, __HIP_MEMORY_SCOPE_AGENT);
}

// deg -> D^{-1/2}; deg >= 1 always because of self loops
__global__ void rsqrt_kernel(float* __restrict__ d, int n) {
  int i = blockIdx.x * blockDim.x + threadIdx.x;
  if (i < n) d[i] = rsqrtf(d[i]);
}

// ---------------------------------------------------------------- W repack
// Pack W[128, COUT] into WMMA B-fragment order so the GEMM inner loop does one
// coalesced b64 load per (k-step, tile):
//   Wr[((kstep*NT + t)*32 + lane)*2 + w] = W[(4*kstep + 2*(lane>>4) + w)*COUT + t*16 + (lane&15)]
template <int COUT>
__global__ void repack_w_kernel(const float* __restrict__ W, float* __restrict__ Wr) {
  constexpr int NT = COUT / 16;
  const int i = blockIdx.x * blockDim.x + threadIdx.x;
  if (i >= 128 * COUT) return;
  const int which = i & 1;
  const int pair  = i >> 1;
  const int lane  = pair & 31;
  const int t     = (pair >> 5) % NT;
  const int ks    = (pair >> 5) / NT;
  const int half  = lane >> 4;
  const int subN  = lane & 15;
  const int krow  = 4 * ks + 2 * half + which;
  Wr[i] = W[(size_t)krow * COUT + t * 16 + subN];
}

// ---------------------------------------------------------------- GEMM (fp32 WMMA)
// C[N, COUT] = A[N,128] @ W[128, COUT] (+ bias). One wave computes 16 rows x COUT.
// Wr is the fragment-packed weight (see repack_w_kernel).
template <int COUT>
__global__ __launch_bounds__(128) void gemm_wmma_kernel(
    const float* __restrict__ A, const v2f* __restrict__ Wr,
    const float* __restrict__ bias, float* __restrict__ C, int N) {
  constexpr int NT = COUT / 16;
  const int wave = threadIdx.x >> 5;
  const int lane = threadIdx.x & 31;
  const int rowBase = (blockIdx.x * 4 + wave) * 16;

  const int subM = lane & 15;   // M (for A) / N (for B,C) within tile
  const int half = lane >> 4;   // selects K pair (A/B) or M upper half (C/D)

  const int aRow = rowBase + subM;
  const int aRowC = (aRow < N) ? aRow : (N - 1);
  const float rowMask = (aRow < N) ? 1.0f : 0.0f;   // keep EXEC all-1s; zero OOB rows
  const float* __restrict__ arow = A + (size_t)aRowC * 128;

  v8f acc[NT] = {};

  for (int ks = 0; ks < 32; ++ks) {
    const int k = ks * 4;
    // A 16x4 f32 fragment: v0 = K = 2*half, v1 = K = 2*half+1  (one b64 load)
    v2f a;
    a.x = arow[k + 2 * half] * rowMask;
    a.y = arow[k + 2 * half + 1] * rowMask;
    const v2f* __restrict__ bp = Wr + (size_t)ks * NT * 32 + lane;
#pragma unroll
    for (int t = 0; t < NT; ++t) {
      v2f b = bp[t * 32];       // single coalesced b64 per fragment
      acc[t] = __builtin_amdgcn_wmma_f32_16x16x4_f32(
          false, a, false, b, (short)0, acc[t], false, false);
    }
  }

#pragma unroll
  for (int t = 0; t < NT; ++t) {
    const int cidx = t * 16 + subM;
    const float bv = bias ? bias[cidx] : 0.0f;
#pragma unroll
    for (int i = 0; i < 8; ++i) {
      const int r = rowBase + half * 8 + i;   // C/D layout: vgpr i -> M=i (+8 upper half)
      if (r < N) C[(size_t)r * COUT + cidx] = acc[t][i] + bv;
    }
  }
}

// ---------------------------------------------------------------- edge aggregation
// out[row] += h[col] * dis[row]*dis[col]; one wave32 per edge, float4 per lane.
__global__ void aggregate_kernel(const float* __restrict__ h,
                                 const int* __restrict__ row, const int* __restrict__ col,
                                 const float* __restrict__ dis,
                                 float* __restrict__ out, int E, int Etot) {
  const int wid  = blockIdx.x * (blockDim.x >> 5) + (threadIdx.x >> 5);
  const int lane = threadIdx.x & 31;
  if (wid >= Etot) return;
  int r, c;
  if (wid < E) { r = row[wid]; c = col[wid]; }
  else         { r = c = wid - E; }          // self loop
  const float nrm = dis[r] * dis[c];
  const float4* __restrict__ src = (const float4*)(h + (size_t)c * 128);
  float4 m = src[lane];                       // 32 lanes x float4 = 128 floats
  float* dst = out + (size_t)r * 128 + lane * 4;
  __hip_atomic_fetch_add(dst + 0, m.x * nrm, __ATOMIC_RELAXED, __HIP_MEMORY_SCOPE_AGENT);
  __hip_atomic_fetch_add(dst + 1, m.y * nrm, __ATOMIC_RELAXED, __HIP_MEMORY_SCOPE_AGENT);
  __hip_atomic_fetch_add(dst + 2, m.z * nrm, __ATOMIC_RELAXED, __HIP_MEMORY_SCOPE_AGENT);
  __hip_atomic_fetch_add(dst + 3, m.w * nrm, __ATOMIC_RELAXED, __HIP_MEMORY_SCOPE_AGENT);
}

// ---------------------------------------------------------------- batchnorm
// stats[0..127]=sum, [128..255]=sumsq, [256..383]=scale, [384..511]=shift
__global__ void bn_stats_kernel(const float* __restrict__ x, float* __restrict__ stats, int N) {
  __shared__ float sdata[512];
  const int tid = threadIdx.x;          // 256 threads = 2 rows x 128 channels
  const int ch = tid & 127;
  float s = 0.f, ss = 0.f;
  const int rowStride = gridDim.x * 2;
  for (int r = blockIdx.x * 2 + (tid >> 7); r < N; r += rowStride) {
    float v = x[(size_t)r * 128 + ch];
    s += v; ss += v * v;
  }
  sdata[tid] = s; sdata[256 + tid] = ss;
  __syncthreads();
  if (tid < 128) {
    s  = sdata[tid] + sdata[tid + 128];
    ss = sdata[256 + tid] + sdata[256 + tid + 128];
    __hip_atomic_fetch_add(&stats[ch],       s,  __ATOMIC_RELAXED, __HIP_MEMORY_SCOPE_AGENT);
    __hip_atomic_fetch_add(&stats[128 + ch], ss, __ATOMIC_RELAXED, __HIP_MEMORY_SCOPE_AGENT);
  }
}

__global__ void bn_finalize_kernel(float* __restrict__ stats,
                                   const float* __restrict__ g, const float* __restrict__ b,
                                   float invN) {
  const int ch = threadIdx.x;   // 128 threads
  const float mean = stats[ch] * invN;
  const float var  = stats[128 + ch] * invN - mean * mean;   // biased variance
  const float scale = g[ch] * rsqrtf(var + BN_EPS);
  stats[256 + ch] = scale;
  stats[384 + ch] = b[ch] - mean * scale;
}

__global__ void bn_apply_relu_kernel(float* __restrict__ x, const float* __restrict__ stats,
                                     long n4 /* N*32 float4s */) {
  const float* __restrict__ scale = stats + 256;
  const float* __restrict__ shift = stats + 384;
  float4* __restrict__ p = (float4*)x;
  long i = (long)blockIdx.x * blockDim.x + threadIdx.x;
  const long stride = (long)gridDim.x * blockDim.x;
  for (; i < n4; i += stride) {
    const int cb = (int)(i & 31) * 4;
    float4 v = p[i];
    v.x = fmaxf(v.x * scale[cb + 0] + shift[cb + 0], 0.f);
    v.y = fmaxf(v.y * scale[cb + 1] + shift[cb + 1], 0.f);
    v.z = fmaxf(v.z * scale[cb + 2] + shift[cb + 2], 0.f);
    v.w = fmaxf(v.w * scale[cb + 3] + shift[cb + 3], 0.f);
    p[i] = v;
  }
}

// ---------------------------------------------------------------- driver

extern "C" void kernel_launch(void* const* d_in, const int* in_sizes, int n_in,
                              void* d_out, int out_size, void* d_ws, size_t ws_size,
                              hipStream_t stream) {
  const float* x   = (const float*)d_in[0];
  const int*   ei  = (const int*)d_in[1];
  const float* w1  = (const float*)d_in[2];
  const float* g1  = (const float*)d_in[3];
  const float* b1  = (const float*)d_in[4];
  const float* w2  = (const float*)d_in[5];
  const float* g2  = (const float*)d_in[6];
  const float* b2  = (const float*)d_in[7];
  const float* w3  = (const float*)d_in[8];
  const float* g3  = (const float*)d_in[9];
  const float* b3  = (const float*)d_in[10];
  const float* fcw = (const float*)d_in[11];
  const float* fcb = (const float*)d_in[12];
  float* out = (float*)d_out;

  const int N = in_sizes[0] / 128;
  const int E = in_sizes[1] / 2;
  const int Etot = E + N;
  const int* row = ei;
  const int* col = ei + E;

  // workspace layout
  float* dis   = (float*)d_ws;                 // N
  float* bufA  = dis + N;                      // N*128
  float* bufB  = bufA + (size_t)N * 128;       // N*128
  float* stats = bufB + (size_t)N * 128;       // 512
  float* wrep  = stats + 512;                  // 128*128 (fragment-packed weights)

  // D^{-1/2}
  fill_f32<<<256, 256, 0, stream>>>(dis, 1.0f, N);                 // self-loop degree
  degree_kernel<<<2048, 256, 0, stream>>>(row, dis, E);
  rsqrt_kernel<<<(N + 255) / 256, 256, 0, stream>>>(dis, N);

  const int gemmBlocks = (N + 63) / 64;        // 4 waves x 16 rows per block
  const int aggBlocks  = (Etot + 7) / 8;       // 8 waves per 256-thread block
  const long nElems = (long)N * 128;
  const float invN = 1.0f / (float)N;

  const float* ws3[3] = {w1, w2, w3};
  const float* gs[3] = {g1, g2, g3};
  const float* bs[3] = {b1, b2, b3};

  const float* cur = x;
  for (int l = 0; l < 3; ++l) {
    repack_w_kernel<128><<<(128 * 128 + 255) / 256, 256, 0, stream>>>(ws3[l], wrep);
    gemm_wmma_kernel<128><<<gemmBlocks, 128, 0, stream>>>(cur, (const v2f*)wrep, nullptr, bufA, N);
    fill_f32<<<512, 256, 0, stream>>>(bufB, 0.0f, nElems);
    aggregate_kernel<<<aggBlocks, 256, 0, stream>>>(bufA, row, col, dis, bufB, E, Etot);
    fill_f32<<<1, 256, 0, stream>>>(stats, 0.0f, 256);
    bn_stats_kernel<<<1024, 256, 0, stream>>>(bufB, stats, N);
    bn_finalize_kernel<<<1, 128, 0, stream>>>(stats, gs[l], bs[l], invN);
    bn_apply_relu_kernel<<<512, 256, 0, stream>>>(bufB, stats, (long)N * 32);
    cur = bufB;
  }

  repack_w_kernel<64><<<(128 * 64 + 255) / 256, 256, 0, stream>>>(fcw, wrep);
  gemm_wmma_kernel<64><<<gemmBlocks, 128, 0, stream>>>(bufB, (const v2f*)wrep, fcb, out, N);
}